// DistanceNetwork_893353197965
// MI455X (gfx1250) — compile-verified
//
#include <hip/hip_runtime.h>

typedef float v2f __attribute__((ext_vector_type(2)));
typedef float v8f __attribute__((ext_vector_type(8)));

// One block per batch. Each wave (32 lanes) owns a 16-row tile of support.
// WMMA f32 16x16x4: A = 16 support rows x 4 K, B = query broadcast into all
// 16 columns -> every column of D holds the 16 dots. Norms via per-lane FMA
// over the same A fragment + one xor-16 shuffle reduction.
__global__ __launch_bounds__(512)
void distance_network_kernel(const float* __restrict__ support,
                             const float* __restrict__ query,
                             float* __restrict__ out,
                             int CK, int D) {
  const int b    = blockIdx.x;
  const int lane = threadIdx.x & 31;
  const int wave = threadIdx.x >> 5;
  const int wavesPerBlock = blockDim.x >> 5;
  const int tilesPerBatch = CK >> 4;      // 16-row tiles
  const int hi = lane >> 4;               // 0 = lanes 0-15, 1 = lanes 16-31
  const int m  = lane & 15;               // row within tile (A fragment M)

  const float* qbase = query + (size_t)b * D + 2 * hi;

  for (int tile = wave; tile < tilesPerBatch; tile += wavesPerBlock) {
    const int rowbase = tile << 4;
    // A-fragment address per ISA layout: lane holds (K, K+1), K = k0 + 2*hi
    const float* ap = support + ((size_t)b * CK + rowbase + m) * (size_t)D + 2 * hi;
    const float* qp = qbase;

    v8f acc = {};
    float ns = 0.0f;

#pragma unroll 4
    for (int k0 = 0; k0 < D; k0 += 4) {
      v2f a  = *(const v2f*)ap;  ap += 4;   // support fragment (8B aligned)
      v2f qv = *(const v2f*)qp;  qp += 4;   // query broadcast fragment
      acc = __builtin_amdgcn_wmma_f32_16x16x4_f32(
          /*neg_a=*/false, a, /*neg_b=*/false, qv,
          /*c_mod=*/(short)0, acc, /*reuse_a=*/false, /*reuse_b=*/false);
      ns = fmaf(a.x, a.x, ns);
      ns = fmaf(a.y, a.y, ns);
    }

    // Full squared norm of row m: this lane's half-K partial + other half.
    float sq  = ns + __shfl_xor(ns, 16);
    float inv = rsqrtf(fmaxf(sq, 1e-10f));   // rsqrt(clip(sq, EPS))

    // acc[r] on lanes 0-15 = dot(row r), on lanes 16-31 = dot(row 8+r)
    // (all 16 columns identical, so any lane in the half works).
    float myval = 0.0f;
#pragma unroll
    for (int r = 0; r < 8; ++r) {
      float sLo = __shfl(inv, r);        // inv_mag for row r   (from lane r)
      float sHi = __shfl(inv, 8 + r);    // inv_mag for row 8+r (from lane 8+r)
      float res = acc[r] * (hi ? sHi : sLo);
      if ((lane & 7) == r) myval = res;  // lane L<8 -> row L; lane 16+L -> row 8+L
    }
    if ((lane & 15) < 8) {
      int row = (lane & 7) + (hi << 3);  // lanes 0-7 -> rows 0-7, 16-23 -> 8-15
      out[(size_t)b * CK + rowbase + row] = myval;  // coalesced 64B store
    }
  }
}

extern "C" void kernel_launch(void* const* d_in, const int* in_sizes, int n_in,
                              void* d_out, int out_size, void* d_ws, size_t ws_size,
                              hipStream_t stream) {
  (void)n_in; (void)d_ws; (void)ws_size;
  const float* support = (const float*)d_in[0];   // [B, CK, D] f32
  const float* query   = (const float*)d_in[1];   // [B, D]     f32
  float* out = (float*)d_out;                      // [B, CK]    f32

  // Recover shapes: in_sizes[0] = B*CK*D, in_sizes[1] = B*D, out_size = B*CK
  const int CK = in_sizes[0] / in_sizes[1];        // 256
  const int B  = out_size / CK;                    // 2048
  const int D  = in_sizes[1] / B;                  // 512

  int tiles = CK / 16;                             // 16
  int wavesPerBlock = tiles < 16 ? tiles : 16;
  dim3 block(32 * wavesPerBlock);                  // 512 threads = 16 wave32
  dim3 grid(B);                                    // 2048 blocks

  distance_network_kernel<<<grid, block, 0, stream>>>(support, query, out, CK, D);
}